// GATConvDGL_7370163880452
// MI455X (gfx1250) — compile-verified
//
#include <hip/hip_runtime.h>

#define N_NODES 100000
#define N_EDGES 1600000
#define IN_F    128
#define OUT_F   64
#define HEADS   4
#define OH      (OUT_F * HEADS)   // 256
#define NEG_SLOPE 0.2f

typedef float v2f __attribute__((ext_vector_type(2)));
typedef float v8f __attribute__((ext_vector_type(8)));

// ------------------------------------------------------------------
// P0: zero the accumulation buffers (out, denom, segment-max keys)
// ------------------------------------------------------------------
__global__ void init_kernel(float* __restrict__ out, float* __restrict__ denom,
                            unsigned* __restrict__ m_u) {
  int idx = blockIdx.x * blockDim.x + threadIdx.x;
  if (idx < N_NODES * OH) out[idx] = 0.0f;
  if (idx < N_NODES * HEADS) { denom[idx] = 0.0f; m_u[idx] = 0u; }
}

// ------------------------------------------------------------------
// P1: Zp = Z @ W^T + bias via V_WMMA_F32_16X16X4_F32.
// Block = 256 threads = 8 wave32s; block computes 16 rows x 256 cols.
// A (16x4 f32) layout: lanes 0-15 M=0..15 {K=k,k+1}; lanes 16-31 {K=k+2,k+3}
// B (4x16 f32) layout: mirror (N = lane%16, K = khalf + {0,1})
// D (16x16 f32): VGPR v, lanes 0-15 -> M=v; lanes 16-31 -> M=v+8; N=lane%16
// ------------------------------------------------------------------
__global__ __launch_bounds__(256) void gemm_kernel(const float* __restrict__ Z,
                                                   const float* __restrict__ W,
                                                   const float* __restrict__ bias,
                                                   float* __restrict__ Zp) {
  __shared__ __align__(16) float sZ[16 * 132];  // padded stride: conflict-free ds reads
  const int tid = threadIdx.x;
  const int rowbase = blockIdx.x * 16;          // N divisible by 16 (100000/16 = 6250)

  // cooperative, fully coalesced load of the contiguous 16x128 Z tile
  const float* zsrc = Z + (size_t)rowbase * IN_F;
  for (int i = tid; i < 16 * IN_F; i += 256)
    sZ[(i / IN_F) * 132 + (i % IN_F)] = zsrc[i];
  __syncthreads();

  const int wave  = tid >> 5;
  const int lane  = tid & 31;
  const int mrow  = lane & 15;          // M for A, N for B/D
  const int khalf = (lane >> 4) << 1;   // 0 or 2

  for (int ctp = 0; ctp < 2; ++ctp) {
    const int colbase = (wave + ctp * 8) * 16;
    v8f acc = {0.f, 0.f, 0.f, 0.f, 0.f, 0.f, 0.f, 0.f};
    const float* arow = sZ + mrow * 132 + khalf;
    const float* wrow = W + (size_t)(colbase + mrow) * IN_F + khalf;
    for (int k = 0; k < IN_F; k += 4) {
      float2 az = *(const float2*)(arow + k);   // ds_load_b64, conflict-free
      float2 bz = *(const float2*)(wrow + k);   // global_load_b64, L2-resident W
      v2f av; av.x = az.x; av.y = az.y;
      v2f bv; bv.x = bz.x; bv.y = bz.y;
      acc = __builtin_amdgcn_wmma_f32_16x16x4_f32(
          /*neg_a=*/false, av, /*neg_b=*/false, bv,
          /*c_mod=*/(short)0, acc, /*reuse_a=*/false, /*reuse_b=*/false);
    }
    const int ncol = colbase + mrow;
    const float bb = bias[ncol];
    const int rowoff = (lane >> 4) * 8;
#pragma unroll
    for (int v = 0; v < 8; ++v)
      Zp[(size_t)(rowbase + rowoff + v) * OH + ncol] = acc[v] + bb;
  }
}

// ------------------------------------------------------------------
// P2: e_l[n,h] = sum_o Zp[n,o,h]*a_l[o,h]; same for e_r. One thread/node.
// Heads pack perfectly into float4 lanes.
// ------------------------------------------------------------------
__global__ void logits_kernel(const float* __restrict__ Zp,
                              const float* __restrict__ a_l,
                              const float* __restrict__ a_r,
                              float* __restrict__ e_l, float* __restrict__ e_r) {
  int n = blockIdx.x * blockDim.x + threadIdx.x;
  if (n >= N_NODES) return;
  const float4* z4  = (const float4*)(Zp + (size_t)n * OH);
  const float4* al4 = (const float4*)a_l;
  const float4* ar4 = (const float4*)a_r;
  float4 sl = {0, 0, 0, 0}, sr = {0, 0, 0, 0};
  for (int o = 0; o < OUT_F; ++o) {
    float4 z = z4[o], al = al4[o], ar = ar4[o];
    sl.x += z.x * al.x; sl.y += z.y * al.y; sl.z += z.z * al.z; sl.w += z.w * al.w;
    sr.x += z.x * ar.x; sr.y += z.y * ar.y; sr.z += z.z * ar.z; sr.w += z.w * ar.w;
  }
  ((float4*)e_l)[n] = sl;
  ((float4*)e_r)[n] = sr;
}

// monotonic float<->uint mapping for atomicMax-based segment max
__device__ __forceinline__ unsigned ord_u32(float f) {
  unsigned u = __float_as_uint(f);
  return (u & 0x80000000u) ? ~u : (u | 0x80000000u);
}
__device__ __forceinline__ float from_ord(unsigned u) {
  return (u & 0x80000000u) ? __uint_as_float(u & 0x7fffffffu) : __uint_as_float(~u);
}

// ------------------------------------------------------------------
// P3: edge scores + LeakyReLU, store [E,4], segment max by row (dest)
// ------------------------------------------------------------------
__global__ void edge_score_kernel(const int* __restrict__ row, const int* __restrict__ col,
                                  const float* __restrict__ e_l, const float* __restrict__ e_r,
                                  float* __restrict__ score, unsigned* __restrict__ m_u) {
  int e = blockIdx.x * blockDim.x + threadIdx.x;
  if (e >= N_EDGES) return;
  int r = row[e], c = col[e];
  float4 el = ((const float4*)e_l)[r];
  float4 er = ((const float4*)e_r)[c];
  float4 s;
  s.x = el.x + er.x; s.y = el.y + er.y; s.z = el.z + er.z; s.w = el.w + er.w;
  s.x = s.x > 0.f ? s.x : NEG_SLOPE * s.x;
  s.y = s.y > 0.f ? s.y : NEG_SLOPE * s.y;
  s.z = s.z > 0.f ? s.z : NEG_SLOPE * s.z;
  s.w = s.w > 0.f ? s.w : NEG_SLOPE * s.w;
  ((float4*)score)[e] = s;
  atomicMax(&m_u[r * 4 + 0], ord_u32(s.x));
  atomicMax(&m_u[r * 4 + 1], ord_u32(s.y));
  atomicMax(&m_u[r * 4 + 2], ord_u32(s.z));
  atomicMax(&m_u[r * 4 + 3], ord_u32(s.w));
}

// ------------------------------------------------------------------
// P4: a_exp = exp(a - m[row]); denom = segment_sum(a_exp) by row
// ------------------------------------------------------------------
__global__ void edge_exp_kernel(const int* __restrict__ row, float* __restrict__ score,
                                const unsigned* __restrict__ m_u, float* __restrict__ denom) {
  int e = blockIdx.x * blockDim.x + threadIdx.x;
  if (e >= N_EDGES) return;
  int r = row[e];
  float4 s = ((const float4*)score)[e];
  float4 ex;
  ex.x = expf(s.x - from_ord(m_u[r * 4 + 0]));
  ex.y = expf(s.y - from_ord(m_u[r * 4 + 1]));
  ex.z = expf(s.z - from_ord(m_u[r * 4 + 2]));
  ex.w = expf(s.w - from_ord(m_u[r * 4 + 3]));
  ((float4*)score)[e] = ex;
  atomicAdd(&denom[r * 4 + 0], ex.x);
  atomicAdd(&denom[r * 4 + 1], ex.y);
  atomicAdd(&denom[r * 4 + 2], ex.z);
  atomicAdd(&denom[r * 4 + 3], ex.w);
}

// ------------------------------------------------------------------
// P5: out[row,o,:] += (a_exp/denom[row]) * Zp[col,o,:]
// 64 threads per edge, float4 per thread (heads in vector lanes).
// Zp and out both fit in the 192MB L2 -> gathers/atomics stay on-chip.
// ------------------------------------------------------------------
__global__ __launch_bounds__(256) void aggregate_kernel(const int* __restrict__ row,
                                                        const int* __restrict__ col,
                                                        const float* __restrict__ score,
                                                        const float* __restrict__ denom,
                                                        const float* __restrict__ Zp,
                                                        float* __restrict__ out) {
  int gid = blockIdx.x * blockDim.x + threadIdx.x;
  int e = gid >> 6;        // edge
  int o = gid & 63;        // output channel
  if (e >= N_EDGES) return;
  int r = row[e], c = col[e];
  float4 w = ((const float4*)score)[e];
  float4 d = ((const float4*)denom)[r];
  w.x /= d.x; w.y /= d.y; w.z /= d.z; w.w /= d.w;
  float4 z = ((const float4*)Zp)[(size_t)c * OUT_F + o];   // global_load_b128 gather
  float* op = out + ((size_t)r * OUT_F + o) * 4;
  atomicAdd(op + 0, w.x * z.x);
  atomicAdd(op + 1, w.y * z.y);
  atomicAdd(op + 2, w.z * z.z);
  atomicAdd(op + 3, w.w * z.w);
}

// ------------------------------------------------------------------
extern "C" void kernel_launch(void* const* d_in, const int* in_sizes, int n_in,
                              void* d_out, int out_size, void* d_ws, size_t ws_size,
                              hipStream_t stream) {
  (void)in_sizes; (void)n_in; (void)out_size; (void)ws_size;
  const float* Z    = (const float*)d_in[0];
  const int*   row  = (const int*)d_in[1];
  const int*   col  = (const int*)d_in[2];
  const float* W    = (const float*)d_in[3];
  const float* bias = (const float*)d_in[4];
  const float* a_l  = (const float*)d_in[5];
  const float* a_r  = (const float*)d_in[6];
  float* out = (float*)d_out;

  // workspace layout (all regions 16B aligned): ~134.4 MB total
  float*    Zp    = (float*)d_ws;                              // N*256
  float*    e_l   = Zp + (size_t)N_NODES * OH;                 // N*4
  float*    e_r   = e_l + (size_t)N_NODES * HEADS;             // N*4
  float*    denom = e_r + (size_t)N_NODES * HEADS;             // N*4
  unsigned* m_u   = (unsigned*)(denom + (size_t)N_NODES * HEADS); // N*4
  float*    score = (float*)(m_u + (size_t)N_NODES * HEADS);   // E*4

  init_kernel<<<(N_NODES * OH + 255) / 256, 256, 0, stream>>>(out, denom, m_u);
  gemm_kernel<<<N_NODES / 16, 256, 0, stream>>>(Z, W, bias, Zp);
  logits_kernel<<<(N_NODES + 255) / 256, 256, 0, stream>>>(Zp, a_l, a_r, e_l, e_r);
  edge_score_kernel<<<(N_EDGES + 255) / 256, 256, 0, stream>>>(row, col, e_l, e_r, score, m_u);
  edge_exp_kernel<<<(N_EDGES + 255) / 256, 256, 0, stream>>>(row, score, m_u, denom);
  aggregate_kernel<<<(unsigned)((size_t)N_EDGES * 64 / 256), 256, 0, stream>>>(
      row, col, score, denom, Zp, out);
}